// MultiHeadTopKAttention_49804440764769
// MI455X (gfx1250) — compile-verified
//
#include <hip/hip_runtime.h>
#include <math.h>

typedef __attribute__((ext_vector_type(2))) float v2f;
typedef __attribute__((ext_vector_type(4))) float v4f;
typedef __attribute__((ext_vector_type(8))) float v8f;
typedef __attribute__((ext_vector_type(4))) unsigned int u32x4;
typedef __attribute__((ext_vector_type(4))) int i32x4;
typedef __attribute__((ext_vector_type(8))) int i32x8;

#define LSEQ 2048
#define DIN 64
#define HN 4
#define HD 16
#define KSEL 64
#define TILE_ROWS 128
#define XPITCH 68                 // padded floats per LDS row (TDM pad: 64 DW + 4 DW)
#define NTILES (LSEQ / TILE_ROWS) // 16

// Issue one TDM load of a 128x64 f32 tile (row-major, stride 64) into LDS with
// hardware padding to a 68-float pitch.  All descriptor fields are uniform.
__device__ __forceinline__ void tdm_load_tile(unsigned int lds_addr, unsigned long long gaddr)
{
    u32x4 g0;
    g0[0] = 1u;                                   // count=1, is_restore=0, gather=0
    g0[1] = lds_addr;                             // LDS byte address
    g0[2] = (unsigned int)(gaddr & 0xffffffffull);        // global_addr[31:0]
    g0[3] = ((unsigned int)(gaddr >> 32) & 0x01ffffffu)   // global_addr[56:32]
            | 0x80000000u;                        // type=2 ("image")
    i32x8 g1;
    g1[0] = (2 << 16)       // data_size = 4B
          | (1 << 20)       // pad_enable
          | (5 << 22)       // pad_interval: 64 DWORDs
          | (3 << 25);      // pad_amount: 4 DWORDs
    g1[1] = (DIN << 16);            // tensor_dim0 = 64  (bits 79:48)
    g1[2] = (TILE_ROWS << 16);      // tensor_dim1 = 128 (bits 111:80)
    g1[3] = (DIN << 16);            // tile_dim0 = 64    (bits 127:112)
    g1[4] = TILE_ROWS;              // tile_dim1 = 128, tile_dim2 = 0
    g1[5] = DIN;                    // tensor_dim0_stride = 64 (bits 207:160)
    g1[6] = 0;
    g1[7] = 0;
    i32x4 z4 = {0, 0, 0, 0};
    i32x8 z8 = {0, 0, 0, 0, 0, 0, 0, 0};
    __builtin_amdgcn_tensor_load_to_lds(g0, g1, z4, z4, z8, 0);
}

__global__ __launch_bounds__(256) void topk_attn_kernel(
    const float* __restrict__ target,  // [B,64]
    const float* __restrict__ xseq,    // [B,L,64]
    const int*   __restrict__ mask,    // [B,L]
    const float* __restrict__ Wq,      // [64,64]
    const float* __restrict__ Wh,      // [64,64]
    const float* __restrict__ Wv,      // [64,64]
    const float* __restrict__ Wo,      // [64,64]
    float* __restrict__ out)           // [B,64]
{
    __shared__ __align__(16) float xtile[2][TILE_ROWS * XPITCH]; // 2x34816B; buf0 reused as keys
    __shared__ __align__(16) float scores[HN * LSEQ];            // 32768B
    __shared__ float wt[DIN * 16];                               // w~ padded [j][n]
    __shared__ float qv[DIN];
    __shared__ float tgt[DIN];
    __shared__ int   selidx[HN][KSEL];
    __shared__ float selsc[HN][KSEL];
    __shared__ float attnw[HN][KSEL];
    __shared__ float yv[HN][DIN];
    __shared__ float outv[DIN];
    __shared__ int   selcnt[HN];

    const int tid  = threadIdx.x;
    const int lane = tid & 31;
    const int w    = tid >> 5;
    const int b    = blockIdx.x;
    const size_t xbase = (size_t)b * LSEQ * DIN;

    // ---------------- prologue: q = t @ Wq ; w~[j][h] = sum_d Wh[j][16h+d] q[16h+d]
    if (tid < DIN) tgt[tid] = target[(size_t)b * DIN + tid];
    if (tid < HN)  selcnt[tid] = 0;

    // kick the TDM pipeline for tile 0 while the prologue math runs
    if (w == 0) {
        tdm_load_tile((unsigned int)(size_t)&xtile[0][0],
                      (unsigned long long)(size_t)(xseq + xbase));
    }

    __syncthreads();
    if (tid < DIN) {
        float acc = 0.f;
        for (int j = 0; j < DIN; ++j) acc += tgt[j] * Wq[j * DIN + tid];
        qv[tid] = acc;
    }
    __syncthreads();
    for (int i = 0; i < 4; ++i) {
        int e = tid + i * 256;           // 1024 entries of wt[j][n]
        int j = e >> 4, n = e & 15;
        float acc = 0.f;
        if (n < HN) {
            for (int d = 0; d < HD; ++d)
                acc += Wh[j * DIN + n * HD + d] * qv[n * HD + d];
        }
        wt[e] = acc;                      // cols 4..15 zero-padded
    }
    __syncthreads();

    // per-lane B fragments (4x16 f32 tiles, K-chunks of 4), hoisted for whole batch
    const int half = lane >> 4;
    const int n16  = lane & 15;
    v2f bf[16];
    #pragma unroll
    for (int c = 0; c < 16; ++c) {
        int k0 = 4 * c + 2 * half;
        bf[c][0] = wt[k0 * 16 + n16];
        bf[c][1] = wt[(k0 + 1) * 16 + n16];
    }

    // ---------------- main loop: double-buffered TDM feed + WMMA f32 score GEMM
    for (int t = 0; t < NTILES; ++t) {
        const int cur = t & 1;
        __syncthreads();                  // everyone done reading buf[1-cur]
        if (w == 0) {
            if (t + 1 < NTILES) {
                tdm_load_tile((unsigned int)(size_t)&xtile[1 - cur][0],
                              (unsigned long long)(size_t)(xseq + xbase + (size_t)(t + 1) * TILE_ROWS * DIN));
                __builtin_amdgcn_s_wait_tensorcnt(1);   // tile t landed; t+1 in flight
            } else {
                __builtin_amdgcn_s_wait_tensorcnt(0);
            }
        }
        __syncthreads();                  // buf[cur] ready for all waves

        const int rb = w * 16;            // each wave: 16 rows
        v8f acc = {0.f, 0.f, 0.f, 0.f, 0.f, 0.f, 0.f, 0.f};
        const float* arow = &xtile[cur][(rb + n16) * XPITCH + 2 * half];
        #pragma unroll
        for (int c = 0; c < 16; ++c) {
            v2f a = *(const v2f*)(arow + 4 * c);
            acc = __builtin_amdgcn_wmma_f32_16x16x4_f32(
                false, a, false, bf[c], (short)0, acc, false, false);
        }
        if (n16 < HN) {                   // cols 0..3 = heads
            int lbase = t * TILE_ROWS + rb + 8 * half;
            #pragma unroll
            for (int r = 0; r < 8; ++r)
                scores[n16 * LSEQ + lbase + r] = acc[r] * 0.25f;
        }
    }
    __syncthreads();

    // ---------------- mask + ordered-key conversion (keys alias xtile[0])
    unsigned* keys = (unsigned*)&xtile[0][0];
    for (int i = 0; i < 8; ++i) {
        int l  = tid + i * 256;
        int mk = mask[(size_t)b * LSEQ + l];
        for (int h = 0; h < HN; ++h) {
            float s = scores[h * LSEQ + l];
            if (mk == 0) { s = -1e9f; scores[h * LSEQ + l] = s; }
            unsigned u = __float_as_uint(s);
            u = (u & 0x80000000u) ? ~u : (u | 0x80000000u);
            keys[h * LSEQ + l] = u;
        }
    }
    __syncthreads();

    // ---------------- per-head top-64, softmax, weighted gather (one wave per head)
    if (w < HN) {
        const int h = w;
        const unsigned* kh = keys + h * LSEQ;

        unsigned prefix = 0u;             // radix binary search: 64th largest key
        for (int bit = 31; bit >= 0; --bit) {
            unsigned cand = prefix | (1u << bit);
            int cnt = 0;
            #pragma unroll 4
            for (int j = 0; j < 16; ++j) {
                const uint4 kv = *(const uint4*)(kh + lane * 64 + j * 4);
                cnt += (kv.x >= cand) + (kv.y >= cand) + (kv.z >= cand) + (kv.w >= cand);
            }
            for (int s = 16; s >= 1; s >>= 1) cnt += __shfl_xor(cnt, s, 32);
            if (cnt >= KSEL) prefix = cand;
        }
        const unsigned T = prefix;

        for (int j = 0; j < 64; ++j) {    // strictly greater: always < 64 entries
            int l = lane * 64 + j;
            unsigned kk = kh[l];
            if (kk > T) {
                int pos = atomicAdd(&selcnt[h], 1);
                selidx[h][pos] = l;
                selsc[h][pos]  = scores[h * LSEQ + l];
            }
        }
        for (int j = 0; j < 64; ++j) {    // fill ties at T, capped at 64
            int l = lane * 64 + j;
            unsigned kk = kh[l];
            if (kk == T) {
                int pos = atomicAdd(&selcnt[h], 1);
                if (pos < KSEL) {
                    selidx[h][pos] = l;
                    selsc[h][pos]  = scores[h * LSEQ + l];
                }
            }
        }

        // softmax over the 64 selected scores
        float s0 = selsc[h][lane], s1 = selsc[h][lane + 32];
        float mx = fmaxf(s0, s1);
        for (int s = 16; s >= 1; s >>= 1) mx = fmaxf(mx, __shfl_xor(mx, s, 32));
        float e0 = __expf(s0 - mx), e1 = __expf(s1 - mx);
        float sm = e0 + e1;
        for (int s = 16; s >= 1; s >>= 1) sm += __shfl_xor(sm, s, 32);
        float inv = 1.f / sm;
        attnw[h][lane] = e0 * inv;
        attnw[h][lane + 32] = e1 * inv;

        // y_h = sum_r attn_r * x[idx_r]  (coalesced 8B/lane row reads)
        float a0 = 0.f, a1 = 0.f;
        for (int r = 0; r < KSEL; ++r) {
            float aw = attnw[h][r];
            int   ix = selidx[h][r];
            const v2f xv = *(const v2f*)(xseq + xbase + (size_t)ix * DIN + 2 * lane);
            a0 += aw * xv[0];
            a1 += aw * xv[1];
        }
        yv[h][2 * lane]     = a0;
        yv[h][2 * lane + 1] = a1;
    }
    __syncthreads();

    // ---------------- out[c] = sum_j yv[c>>4][j] * Wv[j][c] ; res = out @ Wo
    if (tid < DIN) {
        const int h = tid >> 4;
        float acc = 0.f;
        for (int j = 0; j < DIN; ++j) acc += yv[h][j] * Wv[j * DIN + tid];
        outv[tid] = acc;
    }
    __syncthreads();
    if (tid < DIN) {
        float acc = 0.f;
        for (int j = 0; j < DIN; ++j) acc += outv[j] * Wo[j * DIN + tid];
        out[(size_t)b * DIN + tid] = acc;
    }
}

extern "C" void kernel_launch(void* const* d_in, const int* in_sizes, int n_in,
                              void* d_out, int out_size, void* d_ws, size_t ws_size,
                              hipStream_t stream) {
    const float* target = (const float*)d_in[0];
    const float* xseq   = (const float*)d_in[1];
    const int*   mask   = (const int*)d_in[2];
    const float* Wq     = (const float*)d_in[3];
    const float* Wh     = (const float*)d_in[4];
    const float* Wv     = (const float*)d_in[5];
    const float* Wo     = (const float*)d_in[6];
    float* out          = (float*)d_out;

    const int B = in_sizes[0] / DIN;     // 1024
    topk_attn_kernel<<<B, 256, 0, stream>>>(target, xseq, mask, Wq, Wh, Wv, Wo, out);
}